// FactorizedLinear_39256001085961
// MI455X (gfx1250) — compile-verified
//
#include <hip/hip_runtime.h>

typedef float v2f __attribute__((ext_vector_type(2)));
typedef float v8f __attribute__((ext_vector_type(8)));

#define RANK 512
#define FEAT 4096
#define MTOT 8192               // 4 * 2048
#define BM 128
#define BN 128
#define KT 32
#define LDS_STRIDE (KT + 4)     // 36 dwords: rows stay 16B-aligned, conflict-free b64 reads
#define TILE_DW ((BM + BN) * LDS_STRIDE)   // dwords per double-buffer slot (A tile + B tile)

// ---------------------------------------------------------------------------
// Khatri-rao products, generated directly in the layouts the GEMMs want:
//   in_krT [R=512][I=4096] : in_krT[r][i] = f3[d][r]*f4[e][r]*f5[f][r]
//   out_krs[O=4096][R=512] : out_krs[o][r] = w[r]*f0[a][r]*f1[b][r]*f2[c][r]
// ---------------------------------------------------------------------------
__global__ __launch_bounds__(256) void build_kr_kernel(
    const float* __restrict__ w,
    const float* __restrict__ f0, const float* __restrict__ f1,
    const float* __restrict__ f2, const float* __restrict__ f3,
    const float* __restrict__ f4, const float* __restrict__ f5,
    float* __restrict__ in_krT, float* __restrict__ out_krs) {
  const int idx = blockIdx.x * 256 + threadIdx.x;
  const int half = RANK * FEAT;
  if (idx < half) {
    const int r = idx >> 12;
    const int i = idx & (FEAT - 1);
    const int d = i >> 8, e = (i >> 4) & 15, f = i & 15;
    in_krT[idx] = f3[d * RANK + r] * f4[e * RANK + r] * f5[f * RANK + r];
  } else {
    const int j = idx - half;
    const int o = j >> 9;
    const int r = j & (RANK - 1);
    const int a = o >> 8, b = (o >> 4) & 15, c = o & 15;
    out_krs[j] = w[r] * f0[a * RANK + r] * f1[b * RANK + r] * f2[c * RANK + r];
  }
}

// Issue one 128-row x 32-col f32 tile as async global->LDS b128 copies.
// 1024 16B chunks, 8 per thread (128 threads). Tracked by ASYNCcnt.
__device__ __forceinline__ void async_tile_load(const float* __restrict__ gbase,
                                                int ldg, uint32_t lds_base_bytes,
                                                int tid) {
#pragma unroll
  for (int c = 0; c < 8; ++c) {
    const int id = tid + c * 128;
    const int row = id >> 3;
    const int col = (id & 7) * 4;
    const float* g = gbase + (size_t)row * ldg + col;
    const uint32_t l = lds_base_bytes + (uint32_t)(row * LDS_STRIDE + col) * 4u;
    asm volatile("global_load_async_to_lds_b128 %0, %1, off"
                 :: "v"(l), "v"(g) : "memory");
  }
}

// ---------------------------------------------------------------------------
// C[M][N] = A[M][K] * Bt[N][K]^T (+ bias[N]) with V_WMMA_F32_16X16X4_F32.
// 128 threads = 4 waves; block tile 128x128; each wave computes a 64x64
// quadrant = 4x4 WMMA tiles. Double-buffered LDS fed by async copies.
// Requires M%128==0, N%128==0, K%32==0 (true for both calls).
// ---------------------------------------------------------------------------
__global__ __launch_bounds__(128) void gemm_tn_wmma_kernel(
    const float* __restrict__ A, const float* __restrict__ Bt,
    const float* __restrict__ bias, float* __restrict__ C,
    int M, int N, int K) {
  __shared__ __align__(16) float lds[2 * TILE_DW];

  const int tid   = threadIdx.x;
  const int lane  = tid & 31;
  const int wave  = tid >> 5;       // 0..3
  const int waveM = wave >> 1;      // 0..1 -> 64-row slice
  const int waveN = wave & 1;       // 0..1 -> 64-col slice

  const int blockM = blockIdx.y * BM;
  const int blockN = blockIdx.x * BN;

  const uint32_t lds_base = (uint32_t)(uintptr_t)(&lds[0]);

  v8f acc[4][4];
  const v8f vzero = {0.f, 0.f, 0.f, 0.f, 0.f, 0.f, 0.f, 0.f};
#pragma unroll
  for (int i = 0; i < 4; ++i)
#pragma unroll
    for (int j = 0; j < 4; ++j) acc[i][j] = vzero;

  const int lm = lane & 15;           // row/col within a 16x16 fragment
  const int kh = (lane < 16) ? 0 : 2; // K half selected by lane group

  const float* gA = A  + (size_t)blockM * K;
  const float* gB = Bt + (size_t)blockN * K;

  // prefetch tile 0 into buffer 0
  async_tile_load(gA, K, lds_base, tid);
  async_tile_load(gB, K, lds_base + BM * LDS_STRIDE * 4u, tid);

  const int nk = K / KT;
  for (int kt = 0; kt < nk; ++kt) {
    // own async copies done, then block-wide: tile `kt` fully resident
    asm volatile("s_wait_asynccnt 0x0" ::: "memory");
    __syncthreads();

    const int cur = kt & 1;
    if (kt + 1 < nk) {
      const uint32_t nb = lds_base + (uint32_t)((1 - cur) * TILE_DW) * 4u;
      async_tile_load(gA + (kt + 1) * KT, K, nb, tid);
      async_tile_load(gB + (kt + 1) * KT, K, nb + BM * LDS_STRIDE * 4u, tid);
    }

    const float* As = &lds[cur * TILE_DW];
    const float* Bs = As + BM * LDS_STRIDE;

#pragma unroll
    for (int kk = 0; kk < KT; kk += 4) {
      v2f af[4], bf[4];
#pragma unroll
      for (int i = 0; i < 4; ++i)
        af[i] = *reinterpret_cast<const v2f*>(
            &As[(waveM * 64 + i * 16 + lm) * LDS_STRIDE + kk + kh]);
#pragma unroll
      for (int j = 0; j < 4; ++j)
        bf[j] = *reinterpret_cast<const v2f*>(
            &Bs[(waveN * 64 + j * 16 + lm) * LDS_STRIDE + kk + kh]);
#pragma unroll
      for (int i = 0; i < 4; ++i)
#pragma unroll
        for (int j = 0; j < 4; ++j)
          acc[i][j] = __builtin_amdgcn_wmma_f32_16x16x4_f32(
              false, af[i], false, bf[j], (short)0, acc[i][j], false, false);
    }
    // no trailing barrier needed: next iteration's wait+barrier orders reuse
  }

  // Epilogue. C/D layout: vgpr v, lane<16 -> (M=v, N=lane); lane>=16 -> (M=v+8, N=lane-16)
  const int rowAdd = (lane < 16) ? 0 : 8;
  const bool hasBias = (bias != nullptr);
#pragma unroll
  for (int j = 0; j < 4; ++j) {
    const int col = blockN + waveN * 64 + j * 16 + lm;
    const float bv = hasBias ? bias[col] : 0.0f;
#pragma unroll
    for (int i = 0; i < 4; ++i) {
      const int row0 = blockM + waveM * 64 + i * 16 + rowAdd;
#pragma unroll
      for (int v = 0; v < 8; ++v) {
        float* p = &C[(size_t)(row0 + v) * N + col];
        const float val = acc[i][j][v] + bv;
        if (hasBias)  // final output y: streamed once, bypass L2 re-use
          __builtin_nontemporal_store(val, p);
        else          // intermediate t: re-read 32x in GEMM2, keep cached
          *p = val;
      }
    }
  }
}

extern "C" void kernel_launch(void* const* d_in, const int* in_sizes, int n_in,
                              void* d_out, int out_size, void* d_ws, size_t ws_size,
                              hipStream_t stream) {
  const float* x    = (const float*)d_in[0];  // [4, 2048, 4096]
  const float* w    = (const float*)d_in[1];  // [512]
  const float* f0   = (const float*)d_in[2];
  const float* f1   = (const float*)d_in[3];
  const float* f2   = (const float*)d_in[4];
  const float* f3   = (const float*)d_in[5];
  const float* f4   = (const float*)d_in[6];
  const float* f5   = (const float*)d_in[7];
  const float* bias = (const float*)d_in[8];  // [4096]
  float* y = (float*)d_out;                   // [8192, 4096]

  float* in_krT  = (float*)d_ws;                       // [512][4096]   8 MB
  float* out_krs = in_krT + (size_t)RANK * FEAT;       // [4096][512]   8 MB
  float* t       = out_krs + (size_t)FEAT * RANK;      // [8192][512]  16 MB
  const size_t need = ((size_t)RANK * FEAT * 2 + (size_t)MTOT * RANK) * sizeof(float);
  if (ws_size < need) return;

  build_kr_kernel<<<(2 * RANK * FEAT) / 256, 256, 0, stream>>>(
      w, f0, f1, f2, f3, f4, f5, in_krT, out_krs);

  // t = x @ in_kr           (M=8192, N=512,  K=4096)
  gemm_tn_wmma_kernel<<<dim3(RANK / BN, MTOT / BM), dim3(128), 0, stream>>>(
      x, in_krT, nullptr, t, MTOT, RANK, FEAT);

  // y = t @ out_krs^T + b   (M=8192, N=4096, K=512)
  gemm_tn_wmma_kernel<<<dim3(FEAT / BN, MTOT / BM), dim3(128), 0, stream>>>(
      t, out_krs, bias, y, MTOT, FEAT, RANK);
}